// Glm4v_moeTextTopkRouter_24275155157563
// MI455X (gfx1250) — compile-verified
//
#include <hip/hip_runtime.h>
#include <hip/hip_bf16.h>
#include <math.h>
#include <stdint.h>

typedef __attribute__((ext_vector_type(2))) float v2f;
typedef __attribute__((ext_vector_type(8))) float v8f;

#define HIDDEN 4096
#define NEXP   128
#define TOPK   8
#define MBLK   64     // tokens per block
#define KC     32     // K-chunk staged in LDS (double buffered)
#define XSTR   36     // padded LDS stride (floats): gcd(36,64)=4 -> conflict-free frags
#define WSTR   36
#define LOGSTR 130    // padded logits stride (floats)

#define XS_FL  (MBLK * XSTR)            // 2304 floats
#define WS_FL  (NEXP * WSTR)            // 4608 floats
#define BUF_FL (XS_FL + WS_FL)          // 6912 floats per buffer
#define SMEM_FLOATS (2 * BUF_FL)        // 13824 floats = 55296 B; logits (8320) aliases

// Issue one K-chunk of async global->LDS copies (tracked by ASYNCcnt).
// X: 64x32 floats = 512 float4 -> 2 per thread; W: 128x32 = 1024 float4 -> 4 per thread.
// 6 async-load instructions per wave per chunk.
__device__ __forceinline__ void stage_async(const float* __restrict__ X,
                                            const float* __restrict__ W,
                                            float* xbuf, float* wbuf,
                                            long rowBase, int kc, int tid)
{
    #pragma unroll
    for (int i = 0; i < 2; ++i) {
        int idx = tid + i * 256;
        int r = idx >> 3, k4 = idx & 7;
        unsigned ldsoff = (unsigned)(uintptr_t)(xbuf + r * XSTR + k4 * 4);
        int goff = (int)(((rowBase + r) * HIDDEN + kc + k4 * 4) * 4); // bytes
        asm volatile("global_load_async_to_lds_b128 %0, %1, %2"
                     :: "v"(ldsoff), "v"(goff), "s"(X) : "memory");
    }
    #pragma unroll
    for (int i = 0; i < 4; ++i) {
        int idx = tid + i * 256;
        int r = idx >> 3, k4 = idx & 7;
        unsigned ldsoff = (unsigned)(uintptr_t)(wbuf + r * WSTR + k4 * 4);
        int goff = (int)(((long)r * HIDDEN + kc + k4 * 4) * 4);       // bytes
        asm volatile("global_load_async_to_lds_b128 %0, %1, %2"
                     :: "v"(ldsoff), "v"(goff), "s"(W) : "memory");
    }
}

__global__ __launch_bounds__(256, 2)
void moe_router_fused(const float* __restrict__ X,
                      const float* __restrict__ W,
                      const float* __restrict__ bias,
                      int* __restrict__ outI,
                      float* __restrict__ outW)
{
    __shared__ float smem[SMEM_FLOATS];
    float* xs0 = smem;
    float* ws0 = smem + XS_FL;
    float* xs1 = smem + BUF_FL;
    float* ws1 = smem + BUF_FL + XS_FL;
    float* logits = smem;                 // aliased after GEMM phase

    const int tid  = threadIdx.x;
    const int lane = tid & 31;
    const int wv   = tid >> 5;            // 0..7 (8 wave32s)
    const int half = lane >> 4;           // 0|1 : K-subgroup per WMMA f32 layout
    const int lm   = lane & 15;           // row/col within tile

    const int m0 = (wv >> 1) * 16;        // wave's token-tile base (0,16,32,48)
    const int n0 = (wv & 1) * 64;         // wave's expert-half base (0 or 64)

    const long rowBase = (long)blockIdx.x * MBLK;

    v8f c0 = {}, c1 = {}, c2 = {}, c3 = {};

    const int NCHUNK = HIDDEN / KC;       // 128

    // prologue: chunk 0 into buffer 0
    stage_async(X, W, xs0, ws0, rowBase, 0, tid);

    for (int c = 0; c < NCHUNK; ++c) {
        float* xcur = (c & 1) ? xs1 : xs0;
        float* wcur = (c & 1) ? ws1 : ws0;

        if (c + 1 < NCHUNK) {
            // issue next chunk into the other buffer (safe: it was last read
            // in iteration c-1, which ended with a barrier)
            stage_async(X, W, (c & 1) ? xs0 : xs1, (c & 1) ? ws0 : ws1,
                        rowBase, (c + 1) * KC, tid);
            // async loads complete in order: <=6 outstanding => chunk c landed
            asm volatile("s_wait_asynccnt 6" ::: "memory");
        } else {
            asm volatile("s_wait_asynccnt 0" ::: "memory");
        }
        __syncthreads();   // chunk c visible from every wave

        #pragma unroll
        for (int ks = 0; ks < KC; ks += 4) {
            // A frag: lane<16 -> (K=ks,ks+1), lane>=16 -> (K=ks+2,ks+3)
            v2f a  = *(const v2f*)(xcur + (m0 + lm) * XSTR + ks + 2 * half);
            v2f b0 = *(const v2f*)(wcur + (n0 +  0 + lm) * WSTR + ks + 2 * half);
            v2f b1 = *(const v2f*)(wcur + (n0 + 16 + lm) * WSTR + ks + 2 * half);
            v2f b2 = *(const v2f*)(wcur + (n0 + 32 + lm) * WSTR + ks + 2 * half);
            v2f b3 = *(const v2f*)(wcur + (n0 + 48 + lm) * WSTR + ks + 2 * half);
            c0 = __builtin_amdgcn_wmma_f32_16x16x4_f32(false, a, false, b0, (short)0, c0, false, false);
            c1 = __builtin_amdgcn_wmma_f32_16x16x4_f32(false, a, false, b1, (short)0, c1, false, false);
            c2 = __builtin_amdgcn_wmma_f32_16x16x4_f32(false, a, false, b2, (short)0, c2, false, false);
            c3 = __builtin_amdgcn_wmma_f32_16x16x4_f32(false, a, false, b3, (short)0, c3, false, false);
        }
        __syncthreads();   // all waves done reading buf before it is re-filled
    }

    // ---- scatter accumulators into LDS logits[64][130] ----
    // C/D layout: VGPR j : lanes 0-15 -> (M=j, N=lm) ; lanes 16-31 -> (M=j+8, N=lm)
    #pragma unroll
    for (int j = 0; j < 8; ++j) {
        int rl = m0 + j + 8 * half;
        logits[rl * LOGSTR + n0 +  0 + lm] = c0[j];
        logits[rl * LOGSTR + n0 + 16 + lm] = c1[j];
        logits[rl * LOGSTR + n0 + 32 + lm] = c2[j];
        logits[rl * LOGSTR + n0 + 48 + lm] = c3[j];
    }
    __syncthreads();

    // ---- top-8 per token; each wave handles 8 rows ----
    const float b0s = bias[lane];
    const float b1s = bias[lane + 32];
    const float b2s = bias[lane + 64];
    const float b3s = bias[lane + 96];

    for (int rr = 0; rr < MBLK / 8; ++rr) {
        int  rl = wv * (MBLK / 8) + rr;
        long t  = rowBase + rl;

        float sig[4], cor[4];
        #pragma unroll
        for (int j = 0; j < 4; ++j) {
            float l = logits[rl * LOGSTR + lane + 32 * j];
            float s = 1.0f / (1.0f + __expf(-l));
            sig[j] = s;
            cor[j] = s + ((j == 0) ? b0s : (j == 1) ? b1s : (j == 2) ? b2s : b3s);
        }

        float sum = 0.0f;
        int   selI[TOPK];
        float selW[TOPK];
        #pragma unroll
        for (int k = 0; k < TOPK; ++k) {
            float bv = cor[0]; float bs = sig[0]; int bi = lane;
            #pragma unroll
            for (int j = 1; j < 4; ++j) {
                int ci = lane + 32 * j;
                bool better = (cor[j] > bv) || (cor[j] == bv && ci < bi);
                bv = better ? cor[j] : bv;
                bs = better ? sig[j] : bs;
                bi = better ? ci     : bi;
            }
            #pragma unroll
            for (int off = 16; off >= 1; off >>= 1) {
                float ov = __shfl_xor(bv, off, 32);
                float os = __shfl_xor(bs, off, 32);
                int   oi = __shfl_xor(bi, off, 32);
                bool better = (ov > bv) || (ov == bv && oi < bi);
                bv = better ? ov : bv;
                bs = better ? os : bs;
                bi = better ? oi : bi;
            }
            selI[k] = bi; selW[k] = bs; sum += bs;
            #pragma unroll
            for (int j = 0; j < 4; ++j)
                if (bi == lane + 32 * j) cor[j] = -INFINITY;  // mask winner
        }

        float inv = 1.0f / (sum + 1e-20f);
        if (lane == 0) {
            #pragma unroll
            for (int k = 0; k < TOPK; ++k) {
                outI[t * TOPK + k] = selI[k];
                outW[t * TOPK + k] = selW[k] * inv;
            }
        }
    }
}

extern "C" void kernel_launch(void* const* d_in, const int* in_sizes, int n_in,
                              void* d_out, int out_size, void* d_ws, size_t ws_size,
                              hipStream_t stream) {
    const float* X    = (const float*)d_in[0];   // [T, 4096]
    const float* W    = (const float*)d_in[1];   // [128, 4096]
    const float* bias = (const float*)d_in[2];   // [128]
    const int tokens  = in_sizes[0] / HIDDEN;    // 16384

    int*   outI = (int*)d_out;                           // [T, 8] int32 bits
    float* outW = (float*)d_out + (size_t)tokens * TOPK; // [T, 8] f32

    dim3 grid(tokens / MBLK), block(256);
    hipLaunchKernelGGL(moe_router_fused, grid, block, 0, stream,
                       X, W, bias, outI, outW);
}